// DynamicEmbedder_20641612825461
// MI455X (gfx1250) — compile-verified
//
#include <hip/hip_runtime.h>

#define NUM_HIGH_NODES 100000
#define D_COMMON 128

typedef __attribute__((ext_vector_type(2))) float v2f;
typedef __attribute__((ext_vector_type(4))) float v4f;
typedef __attribute__((ext_vector_type(8))) float v8f;

// ---------------------------------------------------------------------------
// Kernel 1: wave-aggregated partition of batch positions into high/low lists.
// wave32: one atomic per wave per expert; intra-wave ranks via ballot/popc.
// ---------------------------------------------------------------------------
__global__ __launch_bounds__(256) void partition_kernel(
    const int* __restrict__ ids, int B, unsigned* __restrict__ cnt,
    int* __restrict__ list_high, int* __restrict__ list_low) {
  const int b = blockIdx.x * blockDim.x + threadIdx.x;
  const bool valid = (b < B);
  const int id = valid ? ids[b] : 0;
  const bool is_high = valid && (id < NUM_HIGH_NODES);
  const bool is_low  = valid && (id >= NUM_HIGH_NODES);

  const unsigned long long mh = __ballot(is_high);
  const unsigned long long ml = __ballot(is_low);
  const int lane = threadIdx.x & 31;

  int base_h = 0, base_l = 0;
  if (lane == 0) {
    if (mh) base_h = (int)atomicAdd(&cnt[0], (unsigned)__popcll(mh));
    if (ml) base_l = (int)atomicAdd(&cnt[1], (unsigned)__popcll(ml));
  }
  base_h = __shfl(base_h, 0, 32);
  base_l = __shfl(base_l, 0, 32);

  const unsigned long long lt = lane ? (~0ull >> (64 - lane)) : 0ull;
  if (is_high) list_high[base_h + __popcll(mh & lt)] = b;
  if (is_low)  list_low [base_l + __popcll(ml & lt)] = b;
}

// ---------------------------------------------------------------------------
// Kernel 2/3: per-expert gathered GEMM via V_WMMA_F32_16X16X4_F32.
// Block = 256 threads = 8 waves. Tile: 16 gathered rows x 128 output cols.
// Wave w computes the 16x16 output tile for columns [16w, 16w+16).
// A (16 x D) staged in LDS (padded rows -> conflict-free ds_load_b64).
// B per lane: 2 consecutive floats of W[col, :] (global b64, L2-resident).
// Epilogue: stage D tiles in LDS, then coalesced non-temporal b128 row writes.
// ---------------------------------------------------------------------------
template <int D>
__global__ __launch_bounds__(256) void expert_gemm_kernel(
    const int* __restrict__ ids, const float* __restrict__ emb,
    const float* __restrict__ W, const float* __restrict__ bias,
    const int* __restrict__ list, const unsigned* __restrict__ cnt_ptr,
    int id_offset, float* __restrict__ out) {
  constexpr int PAD = 4;          // A row stride D+4 -> 16 distinct banks
  constexpr int OPAD = 4;         // out-stage row stride 132 (16B aligned)
  __shared__ float sA[16][D + PAD];
  __shared__ float sOut[16][D_COMMON + OPAD];
  __shared__ int sPos[16];        // original batch positions
  __shared__ int sGid[16];        // local embedding row indices

  const int cnt = (int)cnt_ptr[0];
  const int tile = blockIdx.x << 4;
  if (tile >= cnt) return;        // surplus blocks exit immediately

  const int tid = threadIdx.x;
  if (tid < 16) {
    int r = tile + tid;
    if (r > cnt - 1) r = cnt - 1; // clamp tail (stores masked later)
    const int pos = list[r];
    sPos[tid] = pos;
    sGid[tid] = ids[pos] - id_offset;
  }
  __syncthreads();

  // Cooperative gather of the 16 embedding rows into LDS (float4 vectorized).
  constexpr int V4_PER_ROW = D / 4;
  constexpr int TOT_V4 = 16 * V4_PER_ROW;
  for (int j = tid; j < TOT_V4; j += 256) {
    const int row = j / V4_PER_ROW;
    const int c4 = j - row * V4_PER_ROW;
    const float4 v = ((const float4*)(emb + (size_t)sGid[row] * D))[c4];
    float* dst = &sA[row][c4 * 4];
    dst[0] = v.x; dst[1] = v.y; dst[2] = v.z; dst[3] = v.w;
  }
  __syncthreads();

  const int wave = tid >> 5;
  const int lane = tid & 31;
  const int n = lane & 15;        // N (and M for A-fetch) index
  const int half = lane >> 4;     // lane half selects K pair
  const int col = (wave << 4) + n;

  // C/D layout: VGPR r holds (M=r, lanes 0-15) and (M=8+r, lanes 16-31);
  // bias depends only on N -> broadcast one scalar into all 8 accumulators.
  v8f acc;
  const float bv = bias[col];
#pragma unroll
  for (int r = 0; r < 8; ++r) acc[r] = bv;

  const float* __restrict__ wrow = W + (size_t)col * D;  // W[col, :]
#pragma unroll 8
  for (int k0 = 0; k0 < D; k0 += 4) {
    const int k = k0 + (half << 1);
    // A 16x4: lane M=lane&15, VGPR0/1 = K = 2*half, 2*half+1  (ds_load_b64)
    const v2f a = *(const v2f*)&sA[n][k];
    // B 4x16: lane N=lane&15, VGPR0/1 = K = 2*half, 2*half+1  (global b64)
    v2f bm;
    bm.x = wrow[k];
    bm.y = wrow[k + 1];
    acc = __builtin_amdgcn_wmma_f32_16x16x4_f32(false, a, false, bm,
                                                (short)0, acc, false, false);
  }

  // Stage this wave's 16x16 D-tile into LDS. Lane halves write rows r and
  // r+8 (row bases 32 banks apart) across contiguous columns -> conflict-free.
#pragma unroll
  for (int r = 0; r < 8; ++r) {
    sOut[r + (half << 3)][col] = acc[r];
  }
  __syncthreads();

  // Coalesced epilogue: each output row is 128 contiguous floats; write as
  // non-temporal b128 so the 256MB output stream doesn't pollute L2.
  for (int j = tid; j < 16 * (D_COMMON / 4); j += 256) {
    const int row = j >> 5;       // D_COMMON/4 == 32
    const int c4 = j & 31;
    if (tile + row < cnt) {
      const v4f v = *(const v4f*)&sOut[row][c4 * 4];
      __builtin_nontemporal_store(
          v, (v4f*)(out + (size_t)sPos[row] * D_COMMON + c4 * 4));
    }
  }
}

// ---------------------------------------------------------------------------
extern "C" void kernel_launch(void* const* d_in, const int* in_sizes, int n_in,
                              void* d_out, int out_size, void* d_ws,
                              size_t ws_size, hipStream_t stream) {
  const int* node_ids  = (const int*)d_in[0];
  const float* emb_hi  = (const float*)d_in[1];
  const float* emb_lo  = (const float*)d_in[2];
  const float* W_high  = (const float*)d_in[3];
  const float* b_high  = (const float*)d_in[4];
  const float* W_low   = (const float*)d_in[5];
  const float* b_low   = (const float*)d_in[6];
  float* out = (float*)d_out;
  const int B = in_sizes[0];

  // Workspace layout: [0,16) counters; then two position lists of B ints.
  char* ws = (char*)d_ws;
  unsigned* cnt = (unsigned*)ws;
  int* list_high = (int*)(ws + 16);
  int* list_low  = (int*)(ws + 16 + (size_t)B * sizeof(int));

  hipMemsetAsync(cnt, 0, 16, stream);  // graph-capture legal

  const int pblocks = (B + 255) / 256;
  partition_kernel<<<pblocks, 256, 0, stream>>>(node_ids, B, cnt, list_high,
                                                list_low);

  // Worst-case grids; blocks beyond the compacted count exit immediately.
  const int gblocks = (B + 15) / 16;
  expert_gemm_kernel<256><<<gblocks, 256, 0, stream>>>(
      node_ids, emb_hi, W_high, b_high, list_high, &cnt[0], 0, out);
  expert_gemm_kernel<64><<<gblocks, 256, 0, stream>>>(
      node_ids, emb_lo, W_low, b_low, list_low, &cnt[1], NUM_HIGH_NODES, out);
}